// SSMLayerFFTComplex_45372034514977
// MI455X (gfx1250) — compile-verified
//
#include <hip/hip_runtime.h>
#include <hip/hip_bf16.h>

typedef float v2f __attribute__((ext_vector_type(2)));
typedef float v8f __attribute__((ext_vector_type(8)));
typedef int   v4i_vs __attribute__((vector_size(16)));   // matches builtin param type
typedef __attribute__((address_space(1))) v4i_vs* g_v4i_p;
typedef __attribute__((address_space(3))) v4i_vs* l_v4i_p;

#define NBATCH 16
#define CCH    128      // in_ch
#define TLEN   8192
#define NPAIR  128
#define OCH    128
#define LCHUNK 128
#define NCHUNK (TLEN / LCHUNK)   // 64
#define NSUB   (LCHUNK / 16)     // 8
#define GSTR   132               // padded stride for G tile in LDS
#define YSTR   17                // padded stride for ys^T in LDS

#if defined(__HIP_DEVICE_COMPILE__) && defined(__gfx1250__) && \
    __has_builtin(__builtin_amdgcn_global_load_async_to_lds_b128)
#define USE_ASYNC 1
#else
#define USE_ASYNC 0
#endif

__device__ __forceinline__ void wait_async0() {
#if defined(__HIP_DEVICE_COMPILE__) && defined(__gfx1250__)
#if __has_builtin(__builtin_amdgcn_s_wait_asynccnt)
    __builtin_amdgcn_s_wait_asynccnt(0);
#else
    asm volatile("s_wait_asynccnt 0x0" ::: "memory");
#endif
#endif
}

#if USE_ASYNC
__device__ __forceinline__ void async_cp32B(const float* gsrc, float* ldst) {
    // copies 32 bytes: two b128 transfers; IOFFSET applies to both addresses
    __builtin_amdgcn_global_load_async_to_lds_b128(
        (g_v4i_p)(void*)gsrc, (l_v4i_p)(void*)ldst, 0, 0);
    __builtin_amdgcn_global_load_async_to_lds_b128(
        (g_v4i_p)(void*)gsrc, (l_v4i_p)(void*)ldst, 16, 0);
}
#endif

// ---------------------------------------------------------------------------
// Kernel 0: per-pair parameters.  A_d = exp(-softplus(rl) + i*omega),
// factor = (A_d - 1)/lambda (or DT), and A_d^L by 7 complex squarings.
// ws layout: [Ar|Ai|Fr|Fi|ALr|ALi] each 128 floats.
// ---------------------------------------------------------------------------
__global__ void ssm_setup(const float* __restrict__ rl,
                          const float* __restrict__ ro,
                          float* __restrict__ ws) {
    int p = threadIdx.x;
    float x  = rl[p];
    float sp = (x > 20.f) ? x : log1pf(expf(x));
    float re = -sp;
    float om = ro[p];
    float er = expf(re);
    float ar = er * cosf(om);
    float ai = er * sinf(om);
    float m2 = re * re + om * om;
    float fr, fi;
    if (sqrtf(m2) > 1e-6f) {
        float nr = ar - 1.f;
        fr = (nr * re + ai * om) / m2;
        fi = (ai * re - nr * om) / m2;
    } else {
        fr = 1.0f;  // DT
        fi = 0.0f;
    }
    float pr = ar, pi = ai;              // A^L, L = 128 = 2^7
#pragma unroll
    for (int i = 0; i < 7; ++i) {
        float t = pr * pr - pi * pi;
        pi = 2.f * pr * pi;
        pr = t;
    }
    ws[p]        = ar;  ws[128 + p] = ai;
    ws[256 + p]  = fr;  ws[384 + p] = fi;
    ws[512 + p]  = pr;  ws[640 + p] = pi;
}

// ---------------------------------------------------------------------------
// Kernel 1/3: per-(batch, chunk) local work.
//  WRITE_Y == false : local scan from zero init, emit chunk-end states.
//  WRITE_Y == true  : scan from carry-in state, emit y via GEMM2.
// ---------------------------------------------------------------------------
template <bool WRITE_Y>
__global__ __launch_bounds__(256) void ssm_chunk(
    const float* __restrict__ u,     // (B, C, T)
    const float* __restrict__ Bc,    // (C, P)
    const float* __restrict__ Cm,    // (P, O)
    const float* __restrict__ ws,    // params
    const float* __restrict__ carry, // (B, NC, P, 2)  [pass 3]
    float* __restrict__ send,        // (B, NC, P, 2)  [pass 1]
    float* __restrict__ yout)        // (B, O, T)      [pass 3]
{
    __shared__ float a_lds[2][CCH * 16];   // double-buffered u^T subtile, [c][t]
    __shared__ float Gs[16 * GSTR];        // G(t,p) subtile
    __shared__ float ysT[NPAIR * YSTR];    // (2*Re s)^T, [p][t]

    const int bid = blockIdx.x;
    const int b   = bid / NCHUNK;
    const int ch  = bid % NCHUNK;
    const int tid = threadIdx.x;
    const int w   = tid >> 5;       // wave id 0..7
    const int l   = tid & 31;       // lane
    const int m   = l & 15;         // M/N index within tile
    const int hg  = l >> 4;         // lane half-group
    const int n   = 16 * w + m;     // this wave's output column
    const int t0  = ch * LCHUNK;

    // warm L2/WGP$ for this wave's B-operand columns (global_prefetch_b8)
#pragma unroll
    for (int c = l; c < CCH; c += 32) {
        __builtin_prefetch(&Bc[c * NPAIR + 16 * w], 0, 3);
        if constexpr (WRITE_Y) __builtin_prefetch(&Cm[c * OCH + 16 * w], 0, 3);
    }

    // scan-thread private state + params (threads 0..127)
    float sr = 0.f, si = 0.f, par = 0.f, pai = 0.f, pfr = 0.f, pfi = 0.f;
    if (tid < NPAIR) {
        par = ws[tid];        pai = ws[128 + tid];
        pfr = ws[256 + tid];  pfi = ws[384 + tid];
        if constexpr (WRITE_Y) {
            const float* c2 = carry + ((size_t)(b * NCHUNK + ch) * NPAIR + tid) * 2;
            sr = c2[0];  si = c2[1];
        }
    }

    // this thread's slice of the u staging copy: 8 consecutive floats
    const int cc   = tid >> 1;
    const int half = tid & 1;
    const float* usrc = u + (size_t)b * CCH * TLEN + (size_t)cc * TLEN + t0 + half * 8;
    const int loff = cc * 16 + half * 8;

#if USE_ASYNC
    // prologue: kick off async copy of subtile 0 into buffer 0
    async_cp32B(usrc, &a_lds[0][loff]);
#else
    float4 r0 = *(const float4*)(usrc);
    float4 r1 = *(const float4*)(usrc + 4);
#endif

    for (int sub = 0; sub < NSUB; ++sub) {
        const int ts  = t0 + sub * 16;
        const int buf = sub & 1;

#if USE_ASYNC
        wait_async0();              // u[buf] landed in LDS (this wave's copies)
        __syncthreads();            // ... and everyone else's
        if (sub + 1 < NSUB) {       // in-flight through GEMM1/scan/GEMM2
            async_cp32B(usrc + (sub + 1) * 16, &a_lds[buf ^ 1][loff]);
        }
#else
        {   // spill staged registers, then start loading the next subtile
            float* dst = &a_lds[buf][loff];
            *(float4*)(dst)     = r0;
            *(float4*)(dst + 4) = r1;
        }
        __syncthreads();
        if (sub + 1 < NSUB) {
            const float* sn = usrc + (sub + 1) * 16;
            r0 = *(const float4*)(sn);
            r1 = *(const float4*)(sn + 4);
        }
#endif

        // ---- GEMM1: G(16t x 128p) = u^T (16x128) @ Bc (128x128) ----
        v8f acc = {};
#pragma unroll 4
        for (int kk = 0; kk < 32; ++kk) {
            int c0 = kk * 4 + 2 * hg;
            v2f av, bv;
            av.x = a_lds[buf][c0 * 16 + m];
            av.y = a_lds[buf][(c0 + 1) * 16 + m];
            bv.x = Bc[c0 * NPAIR + n];
            bv.y = Bc[(c0 + 1) * NPAIR + n];
            acc = __builtin_amdgcn_wmma_f32_16x16x4_f32(
                false, av, false, bv, (short)0, acc, false, false);
        }
#pragma unroll
        for (int r = 0; r < 8; ++r)
            Gs[(r + 8 * hg) * GSTR + n] = acc[r];
        __syncthreads();

        // ---- sequential complex scan over 16 steps (threads 0..127) ----
        if (tid < NPAIR) {
#pragma unroll
            for (int t = 0; t < 16; ++t) {
                float g  = Gs[t * GSTR + tid];
                float br = g * pfr;
                float bi = g * pfi;
                float nsr = par * sr - pai * si + br;
                float nsi = par * si + pai * sr + bi;
                sr = nsr;  si = nsi;
                if constexpr (WRITE_Y) ysT[tid * YSTR + t] = 2.f * sr;
            }
        }
        __syncthreads();

        // ---- GEMM2: y(16t x 128o) = (2 Re s)(16x128) @ C (128x128) ----
        if constexpr (WRITE_Y) {
            v8f acc2 = {};
#pragma unroll 4
            for (int kk = 0; kk < 32; ++kk) {
                int c0 = kk * 4 + 2 * hg;
                v2f av, bv;
                av.x = ysT[c0 * YSTR + m];
                av.y = ysT[(c0 + 1) * YSTR + m];
                bv.x = Cm[c0 * OCH + n];
                bv.y = Cm[(c0 + 1) * OCH + n];
                acc2 = __builtin_amdgcn_wmma_f32_16x16x4_f32(
                    false, av, false, bv, (short)0, acc2, false, false);
            }
            // D layout: vgpr r, lane l -> t = 8*hg + r (contiguous), o = n
            float* dst = yout + (size_t)b * OCH * TLEN + (size_t)n * TLEN + ts + 8 * hg;
            float4 p0 = { acc2[0], acc2[1], acc2[2], acc2[3] };
            float4 p1 = { acc2[4], acc2[5], acc2[6], acc2[7] };
            *(float4*)(dst)     = p0;
            *(float4*)(dst + 4) = p1;
        }
    }

    if constexpr (!WRITE_Y) {
        if (tid < NPAIR) {
            float* e = send + ((size_t)(b * NCHUNK + ch) * NPAIR + tid) * 2;
            e[0] = sr;  e[1] = si;
        }
    }
}

// ---------------------------------------------------------------------------
// Kernel 2: chunk-carry scan.  g_j = A^L g_{j-1} + e_j ; carry_j = g_{j-1}.
// ---------------------------------------------------------------------------
__global__ void ssm_carry(const float* __restrict__ ws,
                          const float* __restrict__ send,
                          float* __restrict__ carry) {
    int idx = blockIdx.x * blockDim.x + threadIdx.x;  // 0..2047
    int b = idx >> 7;
    int p = idx & 127;
    float alr = ws[512 + p], ali = ws[640 + p];
    float cr = 0.f, ci = 0.f;
    for (int j = 0; j < NCHUNK; ++j) {
        size_t o = ((size_t)(b * NCHUNK + j) * NPAIR + p) * 2;
        carry[o]     = cr;
        carry[o + 1] = ci;
        float er = send[o], ei = send[o + 1];
        float t = alr * cr - ali * ci + er;
        ci = alr * ci + ali * cr + ei;
        cr = t;
    }
}

// ---------------------------------------------------------------------------
extern "C" void kernel_launch(void* const* d_in, const int* in_sizes, int n_in,
                              void* d_out, int out_size, void* d_ws, size_t ws_size,
                              hipStream_t stream) {
    const float* u  = (const float*)d_in[0];
    const float* rl = (const float*)d_in[1];
    const float* ro = (const float*)d_in[2];
    const float* Bc = (const float*)d_in[3];
    const float* Cm = (const float*)d_in[4];
    float* out = (float*)d_out;

    float* ws     = (float*)d_ws;
    float* params = ws;                                   // 768 floats
    float* send   = ws + 768;                             // B*NC*P*2
    float* carry  = send + (size_t)NBATCH * NCHUNK * NPAIR * 2;

    ssm_setup<<<1, 128, 0, stream>>>(rl, ro, params);
    ssm_chunk<false><<<NBATCH * NCHUNK, 256, 0, stream>>>(
        u, Bc, Cm, params, nullptr, send, nullptr);
    ssm_carry<<<(NBATCH * NPAIR) / 256, 256, 0, stream>>>(params, send, carry);
    ssm_chunk<true><<<NBATCH * NCHUNK, 256, 0, stream>>>(
        u, Bc, Cm, params, carry, nullptr, out);
}